// LoTLoss_25735444037877
// MI455X (gfx1250) — compile-verified
//
#include <hip/hip_runtime.h>
#include <stdint.h>

#define H 256
#define W 256
#define DD 128               // contiguous axis
#define TILE 4               // output tile edge in H and W
#define LT (TILE + 2)        // 6 (with halo)
#define LROW DD              // 128 floats per LDS row (full D incl. halo)
#define TILE_ELEMS (LT * LT * LROW)   // 4608 floats = 18 KB per array
#define GAMMA_F 267.52f

typedef unsigned int v4u __attribute__((ext_vector_type(4)));
typedef int          v4i __attribute__((ext_vector_type(4)));
typedef int          v8i __attribute__((ext_vector_type(8)));

// Issue one TDM 3D tile load: global (row-major, D fastest) -> LDS, tile 128 x 6 x 6 f32.
// D# layout per CDNA5 ISA ch.8 (groups 0..3), data_size=4B, type=2, count=1.
__device__ __forceinline__ void tdm_load_tile(const float* gsrc, uint32_t lds_off)
{
    uint64_t ga = (uint64_t)(uintptr_t)gsrc;

    v4u g0;
    g0.x = 1u;                                   // count=1 (valid), user descriptor
    g0.y = lds_off;                              // lds_addr (bytes, wave-relative)
    g0.z = (uint32_t)(ga & 0xFFFFFFFFu);         // global_addr[31:0]
    g0.w = (uint32_t)(ga >> 32) | 0x80000000u;   // global_addr[56:32] | type=2 (bits 127:126 = 10b)

    v8i g1;
    g1[0] = 0x00020000;                 // workgroup_mask=0, data_size=2 (4B)
    g1[1] = (int)(128u << 16);          // tensor_dim0[15:0]=128 (D extent)
    g1[2] = (int)(256u << 16);          // tensor_dim0[31:16]=0 | tensor_dim1[15:0]=256 (W extent)
    g1[3] = (int)(128u << 16);          // tensor_dim1[31:16]=0 | tile_dim0=128
    g1[4] = (int)((unsigned)LT | ((unsigned)LT << 16)); // tile_dim1=6, tile_dim2=6
    g1[5] = 128;                        // tensor_dim0_stride[31:0] = 128 (D)
    g1[6] = (int)0x80000000u;           // t0s[47:32]=0 | tensor_dim1_stride[15:0]=32768 (W*D)
    g1[7] = 0;                          // tensor_dim1_stride[47:16] = 0

    v4i g2; g2[0] = 256; g2[1] = 0; g2[2] = 0; g2[3] = 0;  // tensor_dim2=256 (H); tile_dim3=0
    v4i g3; g3[0] = 0;   g3[1] = 0; g3[2] = 0; g3[3] = 0;

    v8i gx; gx[0]=0; gx[1]=0; gx[2]=0; gx[3]=0; gx[4]=0; gx[5]=0; gx[6]=0; gx[7]=0;

    __builtin_amdgcn_tensor_load_to_lds(g0, g1, g2, g3, gx, 0 /*cpol: RT*/);
}

__global__ void __launch_bounds__(256) fused_stencil_kernel(
    const float* __restrict__ phi, const float* __restrict__ lfs,
    const float* __restrict__ mask, const float* __restrict__ TE,
    const float* __restrict__ B0,   const float* __restrict__ kern,
    float* __restrict__ out_b, float* __restrict__ out_d)
{
    __shared__ float sPH[TILE_ELEMS];   // phi tile, overwritten in place with sin(phi)
    __shared__ float sCO[TILE_ELEMS];   // cos(phi)
    __shared__ float sLF[TILE_ELEMS];   // lfs tile

    const int b   = blockIdx.z;
    int h0 = 1 + TILE * (int)blockIdx.y; if (h0 > H - 1 - TILE) h0 = H - 1 - TILE; // shift last tile in
    int w0 = 1 + TILE * (int)blockIdx.x; if (w0 > W - 1 - TILE) w0 = W - 1 - TILE;
    const int tid = threadIdx.x;

    const size_t planeW = (size_t)W * DD;        // 32768
    const size_t volB   = (size_t)H * planeW;    // 8388608
    const size_t tileBase = (size_t)b * volB + (size_t)(h0 - 1) * planeW + (size_t)(w0 - 1) * DD;

    // One wave issues both TDM loads, waits on TENSORcnt, then the workgroup
    // barrier publishes the LDS tiles to the other 7 waves.
    if (tid < 32) {
        tdm_load_tile(phi + tileBase, (uint32_t)(uintptr_t)(void*)sPH);
        tdm_load_tile(lfs + tileBase, (uint32_t)(uintptr_t)(void*)sLF);
        __builtin_amdgcn_s_wait_tensorcnt(0);
    }
    __syncthreads();

    // Phase 1: sin/cos once per loaded point (7x fewer transcendentals than naive).
    for (int i = tid; i < TILE_ELEMS; i += 256) {
        float p = sPH[i];
        float s, c;
        __sincosf(p, &s, &c);
        sCO[i] = c;
        sPH[i] = s;
    }
    __syncthreads();

    // Stencil taps (XLA conv is cross-correlation; kernel is symmetric per axis anyway).
    const float wHm = kern[4],  wHp = kern[22];   // first spatial dim (H, 256)
    const float wWm = kern[10], wWp = kern[16];   // second spatial dim (W, 256)
    const float wDm = kern[12], wDp = kern[14];   // third spatial dim (D, 128)
    const float wC  = kern[13];

    const float scale = -1.0f / (B0[0] * TE[b] * GAMMA_F);

    // Phase 2: 4 x 4 x 126 interior outputs per block.
    for (int i = tid; i < TILE * TILE * LROW; i += 256) {
        const int x = i & (LROW - 1);          // global d (tile covers full D)
        const int y = (i >> 7) & (TILE - 1);   // w offset in tile
        const int z =  i >> 9;                 // h offset in tile
        if (x == 0 || x == LROW - 1) continue; // boundary handled elsewhere

        const int c = ((z + 1) * LT + (y + 1)) * LROW + x;

        float a_r = wC * sCO[c]
                  + wDm * sCO[c - 1]         + wDp * sCO[c + 1]
                  + wWm * sCO[c - LROW]      + wWp * sCO[c + LROW]
                  + wHm * sCO[c - LT * LROW] + wHp * sCO[c + LT * LROW];
        float a_i = wC * sPH[c]
                  + wDm * sPH[c - 1]         + wDp * sPH[c + 1]
                  + wWm * sPH[c - LROW]      + wWp * sPH[c + LROW]
                  + wHm * sPH[c - LT * LROW] + wHp * sPH[c + LT * LROW];
        float dl  = wC * sLF[c]
                  + wDm * sLF[c - 1]         + wDp * sLF[c + 1]
                  + wWm * sLF[c - LROW]      + wWp * sLF[c + LROW]
                  + wHm * sLF[c - LT * LROW] + wHp * sLF[c + LT * LROW];

        const size_t gidx = (size_t)b * volB + (size_t)(h0 + z) * planeW
                          + (size_t)(w0 + y) * DD + (size_t)x;

        // mask / outputs are strictly streaming (read/written exactly once):
        // non-temporal so they don't evict reusable phi/lfs halo lines.
        const float m = __builtin_nontemporal_load(&mask[gidx]);

        __builtin_nontemporal_store((a_i * sCO[c] - a_r * sPH[c]) * m * scale, &out_b[gidx]);
        __builtin_nontemporal_store(dl * m, &out_d[gidx]);
    }
}

// Zero the six boundary faces of both outputs. Threads map 1:1 onto face
// elements (edges/corners written more than once with the same 0 — benign).
__global__ void __launch_bounds__(256) boundary_zero_kernel(float* __restrict__ out_b,
                                                            float* __restrict__ out_d,
                                                            int B)
{
    const size_t planeW = (size_t)W * DD;
    const size_t volB   = (size_t)H * planeW;

    const size_t n0 = (size_t)B * 2 * H * W;    // d = 0 / DD-1 faces
    const size_t n1 = (size_t)B * 2 * H * DD;   // w = 0 / W-1  faces
    const size_t n2 = (size_t)B * 2 * W * DD;   // h = 0 / H-1  faces

    size_t t = (size_t)blockIdx.x * blockDim.x + threadIdx.x;
    size_t gidx;
    if (t < n0) {
        size_t b    = t / (2 * H * W);
        size_t r    = t % (2 * H * W);
        size_t face = r / (H * W);
        size_t r2   = r % (H * W);
        size_t h    = r2 / W;
        size_t w    = r2 % W;
        gidx = b * volB + h * planeW + w * DD + (face ? (DD - 1) : 0);
    } else if (t < n0 + n1) {
        size_t u    = t - n0;
        size_t b    = u / (2 * H * DD);
        size_t r    = u % (2 * H * DD);
        size_t face = r / (H * DD);
        size_t r2   = r % (H * DD);
        size_t h    = r2 / DD;
        size_t d    = r2 % DD;
        gidx = b * volB + h * planeW + (face ? (W - 1) : 0) * DD + d;
    } else if (t < n0 + n1 + n2) {
        size_t u    = t - n0 - n1;
        size_t b    = u / (2 * W * DD);
        size_t r    = u % (2 * W * DD);
        size_t face = r / (W * DD);
        size_t r2   = r % (W * DD);
        size_t w    = r2 / DD;
        size_t d    = r2 % DD;
        gidx = b * volB + (face ? (H - 1) : 0) * planeW + w * DD + d;
    } else {
        return;
    }
    out_b[gidx] = 0.0f;
    out_d[gidx] = 0.0f;
}

extern "C" void kernel_launch(void* const* d_in, const int* in_sizes, int n_in,
                              void* d_out, int out_size, void* d_ws, size_t ws_size,
                              hipStream_t stream)
{
    const float* phi  = (const float*)d_in[0];
    const float* lfs  = (const float*)d_in[1];
    const float* mask = (const float*)d_in[2];
    const float* TE   = (const float*)d_in[3];
    const float* B0   = (const float*)d_in[4];
    const float* kern = (const float*)d_in[5];
    const int B = in_sizes[3];                 // TE has one entry per batch

    float* out_b = (float*)d_out;
    float* out_d = out_b + (size_t)B * H * W * DD;

    const size_t faces = (size_t)B * 2 * (H * W + H * DD + W * DD);
    const int bt = 256;
    const int bb = (int)((faces + bt - 1) / bt);
    boundary_zero_kernel<<<bb, bt, 0, stream>>>(out_b, out_d, B);

    dim3 grid((W - 2 + TILE - 1) / TILE, (H - 2 + TILE - 1) / TILE, B); // 64 x 64 x B
    fused_stencil_kernel<<<grid, 256, 0, stream>>>(phi, lfs, mask, TE, B0, kern,
                                                   out_b, out_d);
}